// GNNConv_56659208569289
// MI455X (gfx1250) — compile-verified
//
#include <hip/hip_runtime.h>
#include <hip/hip_bf16.h>

#define N_NODES 100000
#define DIM 128
#define EPS 1e-5f

typedef float v2f __attribute__((ext_vector_type(2)));
typedef float v4f __attribute__((ext_vector_type(4)));
typedef float v8f __attribute__((ext_vector_type(8)));

// ---------------------------------------------------------------------------
// Zero-fill (agg buffer) — float4 stores
// ---------------------------------------------------------------------------
__global__ void gnn_zero_kernel(float4* __restrict__ p, int n4) {
    int i = blockIdx.x * blockDim.x + threadIdx.x;
    if (i < n4) p[i] = float4{0.f, 0.f, 0.f, 0.f};
}

// ---------------------------------------------------------------------------
// Scatter-add: agg[dst[e], :] += h[src[e], :]
// One wave (32 lanes) per edge; each lane moves a float4 (4 dims) => 128 dims.
// Row reads are one coalesced 512B burst; adds are global_atomic_add_f32
// (non-returning) hitting the L2-resident agg buffer.
// ---------------------------------------------------------------------------
__global__ __launch_bounds__(256)
void gnn_scatter_kernel(const float* __restrict__ h,
                        const int* __restrict__ src,
                        const int* __restrict__ dst,
                        float* __restrict__ agg, int E) {
    int tid  = blockIdx.x * blockDim.x + threadIdx.x;
    int e    = tid >> 5;
    int quad = (tid & 31) << 2;             // 0,4,...,124
    if (e >= E) return;
    int s = src[e];
    int d = dst[e];
    const float4 v = *(const float4*)(h + (size_t)s * DIM + quad);
    float* p = agg + (size_t)d * DIM + quad;
    atomicAdd(p + 0, v.x);
    atomicAdd(p + 1, v.y);
    atomicAdd(p + 2, v.z);
    atomicAdd(p + 3, v.w);
}

// ---------------------------------------------------------------------------
// Fused GraphConv GEMM + epilogue:
//   out = agg @ W_rel + b_rel + h @ W_root          (K = 256 fused)
//   mode 0: out = LN(relu(out))          (input layer)
//   mode 1: out = LN(relu(out + h))      (middle layers, residual)
//   mode 2: out = out                    (output layer)
// 256 threads = 8 waves; each wave owns a 16x128 output tile via 8 WMMA
// f32 16x16x4 accumulators.
//
// Weights staged in LDS PRE-SWIZZLED for the WMMA B-fragment layout:
//   sW[ ((kb*2 + h)*128 + n)*4 + q ]  holds  W[kb*8 + 2h + (q&1) + 4*(q>>1)][n]
// so one ds_load_b128 per lane yields the contiguous even-aligned B pairs
// for TWO k-steps (quad elems {0,1} -> kstep kb*8, {2,3} -> kstep kb*8+4).
// ---------------------------------------------------------------------------
__global__ __launch_bounds__(256)
void gnn_conv_gemm_kernel(const float* __restrict__ Xa,    // agg   [N,128]
                          const float* __restrict__ Xh,    // h_in  [N,128]
                          const float* __restrict__ Wrel,  // [128,128]
                          const float* __restrict__ Wroot, // [128,128]
                          const float* __restrict__ brel,  // [128]
                          const float* __restrict__ gamma, // [128]
                          const float* __restrict__ beta,  // [128]
                          float* __restrict__ Out,         // [N,128]
                          int N, int mode) {
    __shared__ float sW[2][DIM * DIM];      // 128 KB of the 320 KB WGP LDS

    const int t = threadIdx.x;
    for (int i = t; i < DIM * DIM; i += 256) {
        const int k  = i >> 7;
        const int n  = i & 127;
        const int kb = k >> 3;
        const int r  = k & 7;
        const int hh = (r >> 1) & 1;
        const int q  = (r & 1) | ((r >> 2) << 1);
        const int di = ((((kb << 1) + hh) << 7) + n) * 4 + q;
        sW[0][di] = Wrel[i];
        sW[1][di] = Wroot[i];
    }
    __syncthreads();

    const int wave = t >> 5;                // 0..7
    const int lane = t & 31;
    const int half = lane >> 4;             // 0|1
    const int l16  = lane & 15;
    const size_t tile_m = ((size_t)blockIdx.x * 8 + wave) * 16;
    if (tile_m >= (size_t)N) return;        // wave-uniform; EXEC stays all-1s

    v8f c[8];
    const v8f vzero = {0.f, 0.f, 0.f, 0.f, 0.f, 0.f, 0.f, 0.f};
#pragma unroll
    for (int nt = 0; nt < 8; ++nt) c[nt] = vzero;

    // --- fused K=256 accumulation: mat 0 = agg@Wrel, mat 1 = h@Wroot ---
#pragma unroll
    for (int mat = 0; mat < 2; ++mat) {
        const float* X = mat ? Xh : Xa;
        const v4f* Wq = (const v4f*)sW[mat];
        const float* arow = X + (tile_m + (size_t)l16) * DIM;
        for (int kb = 0; kb < 16; ++kb) {
            const float* ap = arow + kb * 8 + 2 * half;
            v2f a0 = *(const v2f*)(ap);          // K = kb*8 + 2h .. +1
            v2f a1 = *(const v2f*)(ap + 4);      // K = kb*8+4 + 2h .. +1
            const v4f* wrow = Wq + ((kb << 1) + half) * DIM;
            v4f bq[8];
#pragma unroll
            for (int nt = 0; nt < 8; ++nt)       // one ds_load_b128 each
                bq[nt] = wrow[nt * 16 + l16];
#pragma unroll
            for (int nt = 0; nt < 8; ++nt) {     // pass 1: 8 independent chains
                v2f b0 = __builtin_shufflevector(bq[nt], bq[nt], 0, 1);
                c[nt] = __builtin_amdgcn_wmma_f32_16x16x4_f32(
                    false, a0, false, b0, (short)0, c[nt], false, false);
            }
#pragma unroll
            for (int nt = 0; nt < 8; ++nt) {     // pass 2
                v2f b1 = __builtin_shufflevector(bq[nt], bq[nt], 2, 3);
                c[nt] = __builtin_amdgcn_wmma_f32_16x16x4_f32(
                    false, a1, false, b1, (short)0, c[nt], false, false);
            }
        }
    }

    // --- epilogue ---
    float bn[8];
#pragma unroll
    for (int nt = 0; nt < 8; ++nt) bn[nt] = brel[nt * 16 + l16];

    if (mode == 2) {                        // output layer: conv only
#pragma unroll
        for (int v = 0; v < 8; ++v) {
            const size_t r = tile_m + v + 8 * half;
#pragma unroll
            for (int nt = 0; nt < 8; ++nt)
                Out[r * DIM + nt * 16 + l16] = c[nt][v] + bn[nt];
        }
        return;
    }

    float gn[8], btn[8];
#pragma unroll
    for (int nt = 0; nt < 8; ++nt) {
        gn[nt]  = gamma[nt * 16 + l16];
        btn[nt] = beta[nt * 16 + l16];
    }

#pragma unroll
    for (int v = 0; v < 8; ++v) {
        const size_t r = tile_m + v + 8 * half;   // global row of this C slice
        float x[8];
        float s = 0.f, q = 0.f;
#pragma unroll
        for (int nt = 0; nt < 8; ++nt) {
            float val = c[nt][v] + bn[nt];
            if (mode == 1) val += Xh[r * DIM + nt * 16 + l16];  // residual
            val = fmaxf(val, 0.f);                              // ReLU
            x[nt] = val;
            s += val;
            q += val * val;
        }
        // row of 128 values lives in 16 lanes (within this half) x 8 nt:
        // butterfly-reduce across the 16-lane group
#pragma unroll
        for (int off = 1; off < 16; off <<= 1) {
            s += __shfl_xor(s, off, 32);
            q += __shfl_xor(q, off, 32);
        }
        const float mu  = s * (1.f / DIM);
        const float var = q * (1.f / DIM) - mu * mu;
        const float inv = rsqrtf(var + EPS);
#pragma unroll
        for (int nt = 0; nt < 8; ++nt)
            Out[r * DIM + nt * 16 + l16] = (x[nt] - mu) * inv * gn[nt] + btn[nt];
    }
}

// ---------------------------------------------------------------------------
// Host launcher. Inputs (setup_inputs order):
//  0 in_feat [N,128] f32, 1 edge_index [2,E] i32, 2 W_rel [4,128,128] f32,
//  3 b_rel [4,128] f32, 4 W_root [4,128,128] f32, 5 ln_gamma [128], 6 ln_beta [128]
// ---------------------------------------------------------------------------
extern "C" void kernel_launch(void* const* d_in, const int* in_sizes, int n_in,
                              void* d_out, int out_size, void* d_ws, size_t ws_size,
                              hipStream_t stream) {
    const float* in_feat = (const float*)d_in[0];
    const int*   eidx    = (const int*)d_in[1];
    const float* W_rel   = (const float*)d_in[2];
    const float* b_rel   = (const float*)d_in[3];
    const float* W_root  = (const float*)d_in[4];
    const float* gamma   = (const float*)d_in[5];
    const float* beta    = (const float*)d_in[6];

    const int N = in_sizes[0] / DIM;        // 100000
    const int E = in_sizes[1] / 2;          // 1600000
    const int* src = eidx;
    const int* dst = eidx + E;

    const size_t ND = (size_t)N * DIM;
    float* hA  = (float*)d_ws;              // ping
    float* hB  = hA + ND;                   // pong
    float* agg = hB + ND;                   // scatter accumulator

    const float* hin[4]  = {in_feat, hA, hB, hA};
    float*       hout[4] = {hA, hB, hA, (float*)d_out};
    const int    modes[4] = {0, 1, 1, 2};

    const int zeroBlocks    = (int)((ND / 4 + 255) / 256);
    const int scatterBlocks = (int)(((size_t)E * 32 + 255) / 256);
    const int gemmBlocks    = (N + 127) / 128;   // 8 waves x 16 rows per block

    for (int l = 0; l < 4; ++l) {
        gnn_zero_kernel<<<zeroBlocks, 256, 0, stream>>>((float4*)agg, (int)(ND / 4));
        gnn_scatter_kernel<<<scatterBlocks, 256, 0, stream>>>(hin[l], src, dst, agg, E);
        gnn_conv_gemm_kernel<<<gemmBlocks, 256, 0, stream>>>(
            agg, hin[l],
            W_rel + (size_t)l * DIM * DIM,
            W_root + (size_t)l * DIM * DIM,
            b_rel + (size_t)l * DIM,
            gamma, beta, hout[l], N, modes[l]);
    }
}